// GCN_31568009625965
// MI455X (gfx1250) — compile-verified
//
#include <hip/hip_runtime.h>
#include <cstddef>

#define N_NODES   100000
#define N_EDGES   1600000
#define IN_CH     128
#define HID       128
#define HID2      64
#define NUM_CLASSES 10
#define NUM_GRAPHS  128
#define SLOPE     0.01f

typedef __attribute__((ext_vector_type(2))) float v2f;
typedef __attribute__((ext_vector_type(8))) float v8f;

__device__ __forceinline__ void atomAddF(float* p, float v) {
  __hip_atomic_fetch_add(p, v, __ATOMIC_RELAXED, __HIP_MEMORY_SCOPE_AGENT);
}

__global__ void zero_kernel(float* __restrict__ p, int n) {
  int i = blockIdx.x * blockDim.x + threadIdx.x;
  if (i < n) p[i] = 0.0f;
}

__global__ void degree_kernel(const int* __restrict__ dst, float* __restrict__ deg, int n) {
  int i = blockIdx.x * blockDim.x + threadIdx.x;
  if (i < n) atomAddF(&deg[dst[i]], 1.0f);
}

__global__ void dinv_kernel(float* __restrict__ deg, int n) {
  int i = blockIdx.x * blockDim.x + threadIdx.x;
  if (i < n) deg[i] = rsqrtf(deg[i] + 1.0f);   // self-loop included
}

// C[M,N] = A[M,K] @ B[K,N], row-major, one wave per 16x16 tile.
// grid = (M/16, N/16), block = 32. N, K compile-time so all strides are
// immediates and the K-loop fully unrolls into a WMMA accumulate chain.
template <int N, int K>
__global__ void wmma_gemm_f32(const float* __restrict__ A, const float* __restrict__ B,
                              float* __restrict__ C) {
  const int lane = threadIdx.x;          // 0..31
  const int half = lane >> 4;            // 0 or 1
  const int l15  = lane & 15;
  const int m0 = blockIdx.x * 16;
  const int n0 = blockIdx.y * 16;

  const float* arow = A + (size_t)(m0 + l15) * K + 2 * half;  // lane's A row base
  const float* bcol = B + (size_t)2 * half * N + n0 + l15;    // lane's B col base

  v8f acc = {};
  #pragma unroll
  for (int k0 = 0; k0 < K; k0 += 4) {
    // A 16x4 f32: VGPR0 = K=k0 (lanes 0-15) / K=k0+2 (lanes 16-31); VGPR1 = +1
    v2f a, b;
    a.x = arow[k0];
    a.y = arow[k0 + 1];
    // B 4x16 f32: row striped across lanes; VGPR0 = K=k0 / K=k0+2, VGPR1 = +1
    b.x = bcol[(size_t)k0 * N];
    b.y = bcol[(size_t)(k0 + 1) * N];
    acc = __builtin_amdgcn_wmma_f32_16x16x4_f32(
        /*neg_a=*/false, a, /*neg_b=*/false, b,
        /*c_mod=*/(short)0, acc, /*reuse_a=*/false, /*reuse_b=*/false);
  }
  // D 16x16 f32: VGPR r -> row m0 + r + 8*half, col n0 + l15
  float* crow = C + (size_t)(m0 + 8 * half) * N + n0 + l15;
  #pragma unroll
  for (int r = 0; r < 8; ++r) crow[(size_t)r * N] = acc[r];
}

// one wave per edge; each lane moves 4 of 128 features
__global__ void edge_agg_kernel(const float* __restrict__ h, float* __restrict__ agg,
                                const int* __restrict__ src, const int* __restrict__ dst,
                                const float* __restrict__ dinv, int n_edges) {
  int gid = blockIdx.x * blockDim.x + threadIdx.x;
  int e = gid >> 5;
  int lane = gid & 31;
  if (e >= n_edges) return;
  int s = src[e];
  int d = dst[e];
  float norm = dinv[s] * dinv[d];
  const float4 hv = reinterpret_cast<const float4*>(h + (size_t)s * HID)[lane];
  float* out = agg + (size_t)d * HID + lane * 4;
  atomAddF(out + 0, hv.x * norm);
  atomAddF(out + 1, hv.y * norm);
  atomAddF(out + 2, hv.z * norm);
  atomAddF(out + 3, hv.w * norm);
}

// agg = leaky_relu(agg + h * dinv^2 + bias)
__global__ void node_update_kernel(float* __restrict__ agg, const float* __restrict__ h,
                                   const float* __restrict__ dinv,
                                   const float* __restrict__ bias) {
  int idx = blockIdx.x * blockDim.x + threadIdx.x;
  if (idx >= N_NODES * HID) return;
  int node = idx >> 7;
  int f = idx & 127;
  float di = dinv[node];
  float v = agg[idx] + h[idx] * (di * di) + bias[f];
  agg[idx] = v > 0.0f ? v : SLOPE * v;
}

__global__ void pool_sum_kernel(const float* __restrict__ h, const int* __restrict__ batch,
                                float* __restrict__ sums) {
  int idx = blockIdx.x * blockDim.x + threadIdx.x;
  if (idx >= N_NODES * HID) return;
  int node = idx >> 7;
  int f = idx & 127;
  atomAddF(&sums[(size_t)batch[node] * HID + f], h[idx]);
}

__global__ void pool_cnt_kernel(const int* __restrict__ batch, float* __restrict__ cnts) {
  int i = blockIdx.x * blockDim.x + threadIdx.x;
  if (i < N_NODES) atomAddF(&cnts[batch[i]], 1.0f);
}

__global__ void pool_mean_kernel(const float* __restrict__ sums, const float* __restrict__ cnts,
                                 float* __restrict__ g) {
  int idx = blockIdx.x * blockDim.x + threadIdx.x;
  if (idx >= NUM_GRAPHS * HID) return;
  int gi = idx >> 7;
  g[idx] = sums[idx] / fmaxf(cnts[gi], 1.0f);
}

__global__ void bias_lrelu_kernel(float* __restrict__ d, const float* __restrict__ b,
                                  int total, int F) {
  int idx = blockIdx.x * blockDim.x + threadIdx.x;
  if (idx >= total) return;
  float v = d[idx] + b[idx % F];
  d[idx] = v > 0.0f ? v : SLOPE * v;
}

__global__ void final_kernel(const float* __restrict__ g, const float* __restrict__ W,
                             const float* __restrict__ b, float* __restrict__ out) {
  int idx = blockIdx.x * blockDim.x + threadIdx.x;
  if (idx >= NUM_GRAPHS * NUM_CLASSES) return;
  int r = idx / NUM_CLASSES;
  int c = idx % NUM_CLASSES;
  float s = b[c];
  #pragma unroll 8
  for (int k = 0; k < HID2; ++k) s += g[r * HID2 + k] * W[k * NUM_CLASSES + c];
  out[idx] = s;
}

extern "C" void kernel_launch(void* const* d_in, const int* in_sizes, int n_in,
                              void* d_out, int out_size, void* d_ws, size_t ws_size,
                              hipStream_t stream) {
  const float* x     = (const float*)d_in[0];
  const int*   ei    = (const int*)d_in[1];
  const int*   src   = ei;
  const int*   dstp  = ei + N_EDGES;
  const int*   batch = (const int*)d_in[2];
  const float* W1 = (const float*)d_in[3];  const float* b1 = (const float*)d_in[4];
  const float* W2 = (const float*)d_in[5];  const float* b2 = (const float*)d_in[6];
  const float* W3 = (const float*)d_in[7];  const float* b3 = (const float*)d_in[8];
  const float* W4 = (const float*)d_in[9];  const float* b4 = (const float*)d_in[10];
  const float* W5 = (const float*)d_in[11]; const float* b5 = (const float*)d_in[12];
  float* out = (float*)d_out;

  float* hA   = (float*)d_ws;                        // [N_NODES, HID]
  float* hB   = hA + (size_t)N_NODES * HID;          // [N_NODES, HID]
  float* dinv = hB + (size_t)N_NODES * HID;          // [N_NODES] (deg then dinv)
  float* sums = dinv + N_NODES;                      // [NUM_GRAPHS, HID]
  float* cnts = sums + NUM_GRAPHS * HID;             // [NUM_GRAPHS]
  float* g1   = cnts + NUM_GRAPHS;                   // [NUM_GRAPHS, HID]
  float* g2   = g1 + NUM_GRAPHS * HID;               // [NUM_GRAPHS, HID2]
  float* g3   = g2 + NUM_GRAPHS * HID2;              // [NUM_GRAPHS, HID2]

  const int NF = N_NODES * HID;                      // 12.8M
  dim3 blk256(256);
  dim3 gridNF((NF + 255) / 256);
  dim3 gridNodes((N_NODES + 255) / 256);
  dim3 gridEdges((N_EDGES + 255) / 256);
  dim3 gridEdgeWaves((N_EDGES * 32 + 255) / 256);    // one wave per edge

  // degree -> dinv
  zero_kernel<<<gridNodes, blk256, 0, stream>>>(dinv, N_NODES);
  degree_kernel<<<gridEdges, blk256, 0, stream>>>(dstp, dinv, N_EDGES);
  dinv_kernel<<<gridNodes, blk256, 0, stream>>>(dinv, N_NODES);

  // layer 1: h = x @ W1 ; agg ; self-loop + bias + lrelu
  wmma_gemm_f32<HID, IN_CH><<<dim3(N_NODES / 16, HID / 16), dim3(32), 0, stream>>>(x, W1, hA);
  zero_kernel<<<gridNF, blk256, 0, stream>>>(hB, NF);
  edge_agg_kernel<<<gridEdgeWaves, blk256, 0, stream>>>(hA, hB, src, dstp, dinv, N_EDGES);
  node_update_kernel<<<gridNF, blk256, 0, stream>>>(hB, hA, dinv, b1);

  // layer 2
  wmma_gemm_f32<HID, HID><<<dim3(N_NODES / 16, HID / 16), dim3(32), 0, stream>>>(hB, W2, hA);
  zero_kernel<<<gridNF, blk256, 0, stream>>>(hB, NF);
  edge_agg_kernel<<<gridEdgeWaves, blk256, 0, stream>>>(hA, hB, src, dstp, dinv, N_EDGES);
  node_update_kernel<<<gridNF, blk256, 0, stream>>>(hB, hA, dinv, b2);

  // global mean pool
  zero_kernel<<<dim3((NUM_GRAPHS * HID + NUM_GRAPHS + 255) / 256), blk256, 0, stream>>>(
      sums, NUM_GRAPHS * HID + NUM_GRAPHS);  // sums + cnts contiguous
  pool_sum_kernel<<<gridNF, blk256, 0, stream>>>(hB, batch, sums);
  pool_cnt_kernel<<<gridNodes, blk256, 0, stream>>>(batch, cnts);
  pool_mean_kernel<<<dim3((NUM_GRAPHS * HID + 255) / 256), blk256, 0, stream>>>(sums, cnts, g1);

  // MLP head
  wmma_gemm_f32<HID2, HID><<<dim3(NUM_GRAPHS / 16, HID2 / 16), dim3(32), 0, stream>>>(g1, W3, g2);
  bias_lrelu_kernel<<<dim3((NUM_GRAPHS * HID2 + 255) / 256), blk256, 0, stream>>>(
      g2, b3, NUM_GRAPHS * HID2, HID2);
  wmma_gemm_f32<HID2, HID2><<<dim3(NUM_GRAPHS / 16, HID2 / 16), dim3(32), 0, stream>>>(g2, W4, g3);
  bias_lrelu_kernel<<<dim3((NUM_GRAPHS * HID2 + 255) / 256), blk256, 0, stream>>>(
      g3, b4, NUM_GRAPHS * HID2, HID2);
  final_kernel<<<dim3((NUM_GRAPHS * NUM_CLASSES + 255) / 256), blk256, 0, stream>>>(
      g3, W5, b5, out);
}